// QKRoPEKVCacheTestModel_70858370449543
// MI455X (gfx1250) — compile-verified
//
#include <hip/hip_runtime.h>
#include <cstdint>

#define T_TOKENS     16384
#define NUM_HEADS    32
#define NUM_KV_HEADS 8
#define HEAD_DIM     128
#define HALF         64
#define NUM_SLOTS    32768

typedef float fvec2 __attribute__((ext_vector_type(2)));
typedef float fvec4 __attribute__((ext_vector_type(4)));
typedef int   ivec4 __attribute__((ext_vector_type(4)));

// ---------------------------------------------------------------------------
// Kernel 1: bulk cache copy (268 MB) via the CDNA5 async global<->LDS data
// mover (ASYNCcnt-tracked, VGPR-free). Key trick: the ISA adds the single
// instruction IOFFSET to BOTH the LDS and global addresses, so by giving the
// global chunks the same 4096 B slot stride as the LDS staging buffer, all 8
// loads/stores share one base address and differ only in the immediate
// offset. cpol=1 -> TH=NT (don't pollute L2 with a 268 MB stream).
// ---------------------------------------------------------------------------
#if __has_builtin(__builtin_amdgcn_global_load_async_to_lds_b128) && \
    __has_builtin(__builtin_amdgcn_global_store_async_from_lds_b128)
#define USE_ASYNC_LDS 1
#else
#define USE_ASYNC_LDS 0
#endif

#if USE_ASYNC_LDS
typedef __attribute__((address_space(1))) ivec4 as1_ivec4;
typedef __attribute__((address_space(3))) ivec4 as3_ivec4;
#endif

__device__ __forceinline__ void wait_async_zero() {
#if __has_builtin(__builtin_amdgcn_s_wait_asynccnt)
    __builtin_amdgcn_s_wait_asynccnt(0);
#else
    asm volatile("s_wait_asynccnt 0" ::: "memory");
#endif
}

#define ASYNC_SLOTS 8
#define SLOT_STRIDE_CHUNKS 256                 // chunks (float4) per slot
#define SLOT_STRIDE_BYTES  (SLOT_STRIDE_CHUNKS * 16)   // 4096 B
#define BLOCK_CHUNKS (ASYNC_SLOTS * SLOT_STRIDE_CHUNKS) // 2048 chunks/block
#define CPOL_NT 1                               // TH[2:0] = NT

__global__ __launch_bounds__(256) void cache_copy_async_kernel(
    const float* __restrict__ src, float* __restrict__ dst, long n4)
{
    const int  tid       = threadIdx.x;
    const long blockBase = (long)blockIdx.x * BLOCK_CHUNKS;
    const fvec4* __restrict__ s4 = (const fvec4*)src;
    fvec4* __restrict__ d4 = (fvec4*)dst;

#if USE_ASYNC_LDS
    __shared__ __align__(16) fvec4 stage[ASYNC_SLOTS][256]; // 32 KB, slot stride 4096 B

    if (blockBase + BLOCK_CHUNKS <= n4) {
        // Fast path: one base address, 8 immediate-offset async ops each way.
        const fvec4* gsrc = s4 + blockBase + tid;
        fvec4*       gdst = d4 + blockBase + tid;
        void*        lbase = (void*)&stage[0][tid];

#define ASYNC_LD(S)                                                        \
        __builtin_amdgcn_global_load_async_to_lds_b128(                    \
            (as1_ivec4*)(uintptr_t)gsrc,                                   \
            (as3_ivec4*)(uint32_t)(uintptr_t)lbase,                        \
            (S) * SLOT_STRIDE_BYTES, CPOL_NT)
#define ASYNC_ST(S)                                                        \
        __builtin_amdgcn_global_store_async_from_lds_b128(                 \
            (as1_ivec4*)(uintptr_t)gdst,                                   \
            (as3_ivec4*)(uint32_t)(uintptr_t)lbase,                        \
            (S) * SLOT_STRIDE_BYTES, CPOL_NT)

        ASYNC_LD(0); ASYNC_LD(1); ASYNC_LD(2); ASYNC_LD(3);
        ASYNC_LD(4); ASYNC_LD(5); ASYNC_LD(6); ASYNC_LD(7);
        wait_async_zero();
        ASYNC_ST(0); ASYNC_ST(1); ASYNC_ST(2); ASYNC_ST(3);
        ASYNC_ST(4); ASYNC_ST(5); ASYNC_ST(6); ASYNC_ST(7);
        // S_ENDPGM performs an implicit wait-idle; no trailing wait needed.
#undef ASYNC_LD
#undef ASYNC_ST
        return;
    }
#endif
    // Guarded tail / fallback: plain NT vectorized copy.
    #pragma unroll
    for (int s = 0; s < ASYNC_SLOTS; ++s) {
        long i = blockBase + (long)s * SLOT_STRIDE_CHUNKS + tid;
        if (i < n4) {
            fvec4 val = __builtin_nontemporal_load(s4 + i);
            __builtin_nontemporal_store(val, d4 + i);
        }
    }
}

// ---------------------------------------------------------------------------
// Kernel 2: fused NeoX RoPE (q,k) + v passthrough + KV-cache scatter.
// One block per token (256 thr = 8 wave32). cos/sin for the 64 freqs are
// computed once into LDS; each wave owns one 128-wide head row per
// iteration, each lane handling rotation pairs (2*lane, 2*lane+1) as float2.
// All global streams are read-once / write-once -> non-temporal hints.
// ---------------------------------------------------------------------------
__global__ __launch_bounds__(256) void rope_scatter_kernel(
    const float* __restrict__ q, const float* __restrict__ k,
    const float* __restrict__ v,
    const int* __restrict__ positions, const int* __restrict__ slots,
    float* __restrict__ q_out, float* __restrict__ k_out,
    float* __restrict__ v_out,
    float* __restrict__ cache_k, float* __restrict__ cache_v)
{
    const int t   = blockIdx.x;
    const int tid = threadIdx.x;

    __shared__ __align__(16) float cs[HALF];
    __shared__ __align__(16) float sn[HALF];

    const int pos  = positions[t];
    const int slot = slots[t];

    if (tid < HALF) {
        // mirror reference fp32 math: inv = 1 / 10000^(2j/128)
        float e   = (float)tid * (2.0f / (float)HEAD_DIM);
        float inv = 1.0f / powf(10000.0f, e);
        float f   = (float)pos * inv;
        float s, c;
        sincosf(f, &s, &c);
        cs[tid] = c;
        sn[tid] = s;
    }
    __syncthreads();

    const int lane = tid & 31;
    const int wave = tid >> 5;

    const fvec2 c2 = ((const fvec2*)cs)[lane];  // conflict-free ds_load_b64
    const fvec2 s2 = ((const fvec2*)sn)[lane];

    // 40 head-rows per token (32 q + 8 k), 8 waves -> 5 rows each.
    for (int r = wave; r < NUM_HEADS + NUM_KV_HEADS; r += 8) {
        const float* src;
        float* dst0;
        float* dst1 = nullptr;
        if (r < NUM_HEADS) {
            size_t off = ((size_t)t * NUM_HEADS + r) * HEAD_DIM;
            src  = q + off;
            dst0 = q_out + off;
        } else {
            int h = r - NUM_HEADS;
            size_t off = ((size_t)t * NUM_KV_HEADS + h) * HEAD_DIM;
            src  = k + off;
            dst0 = k_out + off;
            dst1 = cache_k + ((size_t)slot * NUM_KV_HEADS + h) * HEAD_DIM;
        }
        fvec2 x1 = __builtin_nontemporal_load((const fvec2*)src + lane);
        fvec2 x2 = __builtin_nontemporal_load((const fvec2*)(src + HALF) + lane);
        fvec2 o1 = x1 * c2 - x2 * s2;
        fvec2 o2 = x2 * c2 + x1 * s2;
        __builtin_nontemporal_store(o1, (fvec2*)dst0 + lane);
        __builtin_nontemporal_store(o2, (fvec2*)(dst0 + HALF) + lane);
        if (dst1) {
            __builtin_nontemporal_store(o1, (fvec2*)dst1 + lane);
            __builtin_nontemporal_store(o2, (fvec2*)(dst1 + HALF) + lane);
        }
    }

    // v passthrough + scatter: 1024 floats/token = 256 threads x float4
    {
        size_t voff = (size_t)t * NUM_KV_HEADS * HEAD_DIM;
        const fvec4* vs = (const fvec4*)(v + voff);
        fvec4 val = __builtin_nontemporal_load(vs + tid);
        __builtin_nontemporal_store(val, (fvec4*)(v_out + voff) + tid);
        __builtin_nontemporal_store(
            val,
            (fvec4*)(cache_v + (size_t)slot * NUM_KV_HEADS * HEAD_DIM) + tid);
    }
}

// ---------------------------------------------------------------------------
extern "C" void kernel_launch(void* const* d_in, const int* in_sizes, int n_in,
                              void* d_out, int out_size, void* d_ws, size_t ws_size,
                              hipStream_t stream) {
    (void)in_sizes; (void)n_in; (void)out_size; (void)d_ws; (void)ws_size;

    const float* q         = (const float*)d_in[0];
    const float* k         = (const float*)d_in[1];
    const float* v         = (const float*)d_in[2];
    const float* kv_cache  = (const float*)d_in[3];
    const int*   positions = (const int*)d_in[4];
    const int*   slots     = (const int*)d_in[5];

    float* out = (float*)d_out;
    const size_t qN = (size_t)T_TOKENS * NUM_HEADS * HEAD_DIM;      // 67,108,864
    const size_t kN = (size_t)T_TOKENS * NUM_KV_HEADS * HEAD_DIM;   // 16,777,216
    const size_t cacheHalf = (size_t)NUM_SLOTS * NUM_KV_HEADS * HEAD_DIM; // 33,554,432

    float* q_out   = out;
    float* k_out   = q_out + qN;
    float* v_out   = k_out + kN;
    float* cache   = v_out + kN;
    float* cache_k = cache;
    float* cache_v = cache + cacheHalf;

    // 1) Copy full input kv_cache into output cache region (async DMA path).
    //    n4 = 16,777,216 float4 chunks = 8192 blocks * 2048 chunks/block.
    long n4 = (long)(2 * cacheHalf / 4);
    int  nblocks = (int)((n4 + BLOCK_CHUNKS - 1) / BLOCK_CHUNKS);   // 8192
    cache_copy_async_kernel<<<nblocks, 256, 0, stream>>>(kv_cache, cache, n4);

    // 2) Fused RoPE + scatter (ordered after the copy on the same stream).
    rope_scatter_kernel<<<T_TOKENS, 256, 0, stream>>>(
        q, k, v, positions, slots, q_out, k_out, v_out, cache_k, cache_v);
}